// FourierFilter_29987461660819
// MI455X (gfx1250) — compile-verified
//
#include <hip/hip_runtime.h>

// ---------------------------------------------------------------------------
// FourierFilter on MI455X (gfx1250):
//   4096-pt FFT per (b,c) as radix-64 x radix-64 Cooley-Tukey; each 64-pt DFT
//   stage is a [64x64] x [64x128(channels)] fp32 GEMM on V_WMMA_F32_16X16X4_F32.
//   Each wave owns 4 column tiles of one 16-row band so A-fragments (from a
//   64-entry (cos,sin) LDS table) are reused 4x. Complex stages use Karatsuba
//   (3 WMMA products instead of 4). Async global->LDS staging; per-(b,c)
//   energy-top-k mask via in-LDS bitonic sort + prefix scan.
// ---------------------------------------------------------------------------

#define BB 32
#define TT 4096
#define CC 128
#define RR 64          // radix
#define FQ 2049        // rfft bins
#define BTC (BB * TT * CC)

typedef float v2f __attribute__((ext_vector_type(2)));
typedef float v8f __attribute__((ext_vector_type(8)));
typedef int   v4i __attribute__((ext_vector_type(4)));

typedef __attribute__((address_space(1))) v4i* gptr4;   // global
typedef __attribute__((address_space(3))) v4i* lptr4;   // LDS

// ---------------- CDNA5 WMMA (fp32 16x16x4), guarded ----------------
__device__ __forceinline__ v8f wmma4(v2f a, v2f b, v8f c) {
#if __has_builtin(__builtin_amdgcn_wmma_f32_16x16x4_f32)
    return __builtin_amdgcn_wmma_f32_16x16x4_f32(false, a, false, b,
                                                 (short)0, c, false, false);
#else
    c[0] += a.x * b.x + a.y * b.y;  // keeps compilation alive; wmma==0 signals
    return c;
#endif
}

// ---------------- async global -> LDS (16B per lane), guarded ----------------
__device__ __forceinline__ void g2l16(void* lds, const void* g) {
#if __has_builtin(__builtin_amdgcn_global_load_async_to_lds_b128)
    __builtin_amdgcn_global_load_async_to_lds_b128(
        (gptr4)(v4i*)g, (lptr4)(v4i*)lds, 0, 0);
#else
    *(float4*)lds = *(const float4*)g;
#endif
}

__device__ __forceinline__ void wait_async0() {
#if __has_builtin(__builtin_amdgcn_s_wait_asynccnt)
    __builtin_amdgcn_s_wait_asynccnt(0);
#else
    asm volatile("s_wait_asynccnt 0" ::: "memory");
#endif
}

// Fill the 64-entry exp(i*2pi*p/64) table: G[2p]=cos, G[2p+1]=sin.
__device__ __forceinline__ void fill_g(float* G, int tid) {
    if (tid < 64) {
        float s, c;
        __sincosf(0.09817477042468103f * (float)tid, &s, &c);
        G[2 * tid] = c;
        G[2 * tid + 1] = s;
    }
}

// A-fragment of the 64-pt DFT factor: element (m,k)=exp(i*2pi*(m*k mod 64)/64).
// Same-p lanes broadcast; b64 pair loads are conflict-free.
__device__ __forceinline__ void afrag(const float* G, int m, int ka,
                                      v2f& fc, v2f& fs) {
    const int p0 = ((m * ka) & 63) << 1;
    const int p1 = ((m * (ka + 1)) & 63) << 1;
    fc.x = G[p0];     fs.x = G[p0 + 1];
    fc.y = G[p1];     fs.y = G[p1 + 1];
}

// Wave -> tile mapping: 8 waves, each owns one 16-row band (tm = w&3) and
// 4 column tiles starting at cb = (w>>2)*64. A-fragments shared across tiles.
#define TILE_MAP()                             \
    const int w = tid >> 5, lane = tid & 31;   \
    const int l = lane & 15, hi = lane >> 4;   \
    const int row0 = (w & 3) << 4;             \
    const int cb = (w >> 2) << 6;              \
    const int m = row0 + l;

// ===========================================================================
// Kernel A: forward stage 1.  WG = (b, n2).
//   Y[k1,n2,c] = sum_n1 x[64*n1+n2, c] * exp(-i*2pi*n1*k1/64),
//   then twiddle by exp(-i*2pi*k1*n2/4096). Layout Y[b][k1][n2][c].
// ===========================================================================
__global__ __launch_bounds__(256) void fwd_stage1(const float* __restrict__ x,
                                                  float* __restrict__ yre,
                                                  float* __restrict__ yim) {
    __shared__ float Xs[RR * CC];  // 32 KB
    __shared__ float G[128];
    const int b = blockIdx.x >> 6, n2 = blockIdx.x & 63;
    const int tid = threadIdx.x;

    fill_g(G, tid);
    const float* src = x + ((size_t)b * TT + n2) * CC;
    for (int i = tid; i < RR * (CC / 4); i += 256) {
        int r = i >> 5, c4 = (i & 31) << 2;
        g2l16(&Xs[r * CC + c4], src + (size_t)r * RR * CC + c4);
    }
    wait_async0();
    __syncthreads();

    TILE_MAP();
    v8f tre[4], tim[4];
#pragma unroll
    for (int j = 0; j < 4; ++j) { tre[j] = (v8f){0,0,0,0,0,0,0,0}; tim[j] = (v8f){0,0,0,0,0,0,0,0}; }

#pragma unroll
    for (int kk = 0; kk < 16; ++kk) {
        const int ka = (kk << 2) + (hi << 1);
        v2f fc, fs;
        afrag(G, m, ka, fc, fs);
#pragma unroll
        for (int j = 0; j < 4; ++j) {
            const int col = cb + (j << 4) + l;
            v2f bx;
            bx.x = Xs[ka * CC + col];
            bx.y = Xs[(ka + 1) * CC + col];
            tre[j] = wmma4(fc, bx, tre[j]);   // Re += cos * x
            tim[j] = wmma4(fs, bx, tim[j]);   // (negated at store: Im = -sum sin*x)
        }
    }
#pragma unroll
    for (int v = 0; v < 8; ++v) {
        const int k1 = row0 + v + (hi << 3);
        const float t = 1.5339807878856412e-3f * (float)(k1 * n2);  // 2pi/4096
        float st, ct;
        __sincosf(t, &st, &ct);
        const size_t orow = ((size_t)b * TT + (size_t)k1 * RR + n2) * CC;
#pragma unroll
        for (int j = 0; j < 4; ++j) {
            const int cch = cb + (j << 4) + l;
            const float re = tre[j][v], im = -tim[j][v];
            yre[orow + cch] = re * ct + im * st;   // * exp(-i t)
            yim[orow + cch] = im * ct - re * st;
        }
    }
}

// ===========================================================================
// Kernel B: forward stage 2.  WG = (b, k1).  Karatsuba complex GEMM:
//   (C - iS)(Br + iBi):  T1=C*Br, T2=S*Bi, T3=(C+S)*(Bi-Br)
//   Re = T1 + T2,  Im = T3 + T1 - T2.
// ===========================================================================
__global__ __launch_bounds__(256) void fwd_stage2(const float* __restrict__ yre,
                                                  const float* __restrict__ yim,
                                                  float* __restrict__ xre,
                                                  float* __restrict__ xim) {
    __shared__ float Yr[RR * CC];  // 32 KB
    __shared__ float Yi[RR * CC];  // 32 KB
    __shared__ float G[128];
    const int b = blockIdx.x >> 6, k1 = blockIdx.x & 63;
    const int tid = threadIdx.x;

    fill_g(G, tid);
    const size_t base = ((size_t)b * TT + (size_t)k1 * RR) * CC;
    for (int i = tid; i < RR * (CC / 4); i += 256) {
        g2l16(&Yr[i * 4], yre + base + (size_t)i * 4);
        g2l16(&Yi[i * 4], yim + base + (size_t)i * 4);
    }
    wait_async0();
    __syncthreads();

    TILE_MAP();
    v8f t1[4], t2[4], t3[4];
#pragma unroll
    for (int j = 0; j < 4; ++j) {
        t1[j] = (v8f){0,0,0,0,0,0,0,0};
        t2[j] = (v8f){0,0,0,0,0,0,0,0};
        t3[j] = (v8f){0,0,0,0,0,0,0,0};
    }

#pragma unroll
    for (int kk = 0; kk < 16; ++kk) {
        const int ka = (kk << 2) + (hi << 1);
        v2f fc, fs;
        afrag(G, m, ka, fc, fs);
        v2f fcs; fcs.x = fc.x + fs.x; fcs.y = fc.y + fs.y;
#pragma unroll
        for (int j = 0; j < 4; ++j) {
            const int col = cb + (j << 4) + l;
            v2f br, bi, bs;
            br.x = Yr[ka * CC + col];
            br.y = Yr[(ka + 1) * CC + col];
            bi.x = Yi[ka * CC + col];
            bi.y = Yi[(ka + 1) * CC + col];
            bs.x = bi.x - br.x; bs.y = bi.y - br.y;
            t1[j] = wmma4(fc, br, t1[j]);
            t2[j] = wmma4(fs, bi, t2[j]);
            t3[j] = wmma4(fcs, bs, t3[j]);
        }
    }
#pragma unroll
    for (int v = 0; v < 8; ++v) {
        const int k2 = row0 + v + (hi << 3);
        const size_t orow = ((size_t)b * TT + (size_t)(RR * k2 + k1)) * CC;
#pragma unroll
        for (int j = 0; j < 4; ++j) {
            const int cch = cb + (j << 4) + l;
            const float a = t1[j][v], bq = t2[j][v];
            xre[orow + cch] = a + bq;
            xim[orow + cch] = t3[j][v] + a - bq;
        }
    }
}

// ===========================================================================
// Kernel M: per-(b,c) energy ranking -> mask[b][k][c] for k in [0,2049).
//   Bitonic sort (2049 padded to 4096) + chunked prefix scan in LDS.
//   Spectrum (128 MB) is resident in the 192 MB L2 -> scattered reads are cheap.
// ===========================================================================
__global__ __launch_bounds__(256) void mask_kernel(const float* __restrict__ xre,
                                                   const float* __restrict__ xim,
                                                   float* __restrict__ maskb) {
    __shared__ float e[4096];
    __shared__ int sidx[4096];
    __shared__ float chunk[256];
    __shared__ float totalE;
    __shared__ int kst;
    const int b = blockIdx.x >> 7, c = blockIdx.x & 127;
    const int tid = threadIdx.x;

    for (int i = tid; i < 4096; i += 256) {
        if (i < FQ) {
            const size_t o = ((size_t)b * TT + i) * CC + c;
            const float re = xre[o], im = xim[o];
            e[i] = re * re + im * im;
            sidx[i] = i;
        } else {
            e[i] = 0.0f;       // energies >= 0, zero-pad sinks to the tail
            sidx[i] = 4096;    // sentinel
        }
    }

    // bitonic sort, descending
    for (int kk = 2; kk <= 4096; kk <<= 1) {
        for (int j = kk >> 1; j > 0; j >>= 1) {
            __syncthreads();
            for (int i = tid; i < 4096; i += 256) {
                const int ixj = i ^ j;
                if (ixj > i) {
                    const bool desc = ((i & kk) == 0);
                    const float ei = e[i], ej = e[ixj];
                    if (desc ? (ei < ej) : (ei > ej)) {
                        e[i] = ej; e[ixj] = ei;
                        const int t = sidx[i]; sidx[i] = sidx[ixj]; sidx[ixj] = t;
                    }
                }
            }
        }
    }
    __syncthreads();

    // chunked prefix scan: find smallest position with cum > 0.8*total
    float local = 0.0f;
    const int cb = tid * 16;
    for (int q = 0; q < 16; ++q) local += e[cb + q];
    chunk[tid] = local;
    __syncthreads();
    if (tid == 0) {
        float run = 0.0f;
        for (int q = 0; q < 256; ++q) { const float t = chunk[q]; chunk[q] = run; run += t; }
        totalE = run;
        kst = 4095;
    }
    __syncthreads();
    const float thresh = 0.8f * totalE;
    float run = chunk[tid];
    int found = 4096;
    for (int q = 0; q < 16; ++q) {
        run += e[cb + q];
        if (run > thresh) { found = cb + q; break; }
    }
    if (found < 4096) atomicMin(&kst, found);
    __syncthreads();

    const int ks = kst;  // positions 0..ks inclusive are the masked (top-k) bins
    for (int i = tid; i < 4096; i += 256) {
        const int f = sidx[i];
        if (f < FQ)
            maskb[((size_t)b * FQ + f) * CC + c] = (i <= ks) ? 0.0f : 1.0f;
    }
}

// ===========================================================================
// Kernel C: inverse stage 1 (mask at load).  WG = (b, r).  Karatsuba:
//   (C + iS)(Br + iBi):  T1=C*Br, T2=S*Bi, T3=(C+S)*(Br+Bi)
//   Re = T1 - T2,  Im = T3 - T1 - T2.   Then twiddle exp(+i*2pi*m2*r/4096).
// ===========================================================================
__global__ __launch_bounds__(256) void inv_stage1(const float* __restrict__ xre,
                                                  const float* __restrict__ xim,
                                                  const float* __restrict__ maskb,
                                                  float* __restrict__ ure,
                                                  float* __restrict__ uim) {
    __shared__ float Yr[RR * CC];
    __shared__ float Yi[RR * CC];
    __shared__ float G[128];
    const int b = blockIdx.x >> 6, r = blockIdx.x & 63;
    const int tid = threadIdx.x;

    fill_g(G, tid);
    for (int i = tid; i < RR * CC; i += 256) {
        const int a = i >> 7, c = i & 127;
        const int k = RR * a + r;
        const int kk = (k <= 2048) ? k : (4096 - k);  // Hermitian mirror of mask
        const float mm = maskb[((size_t)b * FQ + kk) * CC + c];
        const size_t o = ((size_t)b * TT + k) * CC + c;
        Yr[i] = xre[o] * mm;
        Yi[i] = xim[o] * mm;
    }
    __syncthreads();

    TILE_MAP();
    v8f t1[4], t2[4], t3[4];
#pragma unroll
    for (int j = 0; j < 4; ++j) {
        t1[j] = (v8f){0,0,0,0,0,0,0,0};
        t2[j] = (v8f){0,0,0,0,0,0,0,0};
        t3[j] = (v8f){0,0,0,0,0,0,0,0};
    }

#pragma unroll
    for (int kk = 0; kk < 16; ++kk) {
        const int ka = (kk << 2) + (hi << 1);
        v2f fc, fs;
        afrag(G, m, ka, fc, fs);
        v2f fcs; fcs.x = fc.x + fs.x; fcs.y = fc.y + fs.y;
#pragma unroll
        for (int j = 0; j < 4; ++j) {
            const int col = cb + (j << 4) + l;
            v2f br, bi, bs;
            br.x = Yr[ka * CC + col];
            br.y = Yr[(ka + 1) * CC + col];
            bi.x = Yi[ka * CC + col];
            bi.y = Yi[(ka + 1) * CC + col];
            bs.x = br.x + bi.x; bs.y = br.y + bi.y;
            t1[j] = wmma4(fc, br, t1[j]);
            t2[j] = wmma4(fs, bi, t2[j]);
            t3[j] = wmma4(fcs, bs, t3[j]);
        }
    }
#pragma unroll
    for (int v = 0; v < 8; ++v) {
        const int m2 = row0 + v + (hi << 3);
        const float t = 1.5339807878856412e-3f * (float)(m2 * r);
        float st, ct;
        __sincosf(t, &st, &ct);  // * exp(+i t)
        const size_t orow = ((size_t)b * TT + (size_t)m2 * RR + r) * CC;
#pragma unroll
        for (int j = 0; j < 4; ++j) {
            const int cch = cb + (j << 4) + l;
            const float a = t1[j][v], bq = t2[j][v];
            const float re = a - bq;
            const float im = t3[j][v] - a - bq;
            ure[orow + cch] = re * ct - im * st;
            uim[orow + cch] = re * st + im * ct;
        }
    }
}

// ===========================================================================
// Kernel D: inverse stage 2 + outputs.  WG = (b, m2).
//   x_var = (1/4096) * ( sum cos*Ur - sum sin*Ui );  x_inv = x - x_var.
// ===========================================================================
__global__ __launch_bounds__(256) void inv_stage2(const float* __restrict__ ure,
                                                  const float* __restrict__ uim,
                                                  const float* __restrict__ x,
                                                  float* __restrict__ xvar,
                                                  float* __restrict__ xinv) {
    __shared__ float Yr[RR * CC];
    __shared__ float Yi[RR * CC];
    __shared__ float G[128];
    const int b = blockIdx.x >> 6, m2 = blockIdx.x & 63;
    const int tid = threadIdx.x;

    fill_g(G, tid);
    const size_t base = ((size_t)b * TT + (size_t)m2 * RR) * CC;
    for (int i = tid; i < RR * (CC / 4); i += 256) {
        g2l16(&Yr[i * 4], ure + base + (size_t)i * 4);
        g2l16(&Yi[i * 4], uim + base + (size_t)i * 4);
    }
    wait_async0();
    __syncthreads();

    TILE_MAP();
    const float invT = 2.44140625e-4f;  // 1/4096
    v8f t1[4], t2[4];
#pragma unroll
    for (int j = 0; j < 4; ++j) { t1[j] = (v8f){0,0,0,0,0,0,0,0}; t2[j] = (v8f){0,0,0,0,0,0,0,0}; }

#pragma unroll
    for (int kk = 0; kk < 16; ++kk) {
        const int ka = (kk << 2) + (hi << 1);
        v2f fc, fs;
        afrag(G, m, ka, fc, fs);
#pragma unroll
        for (int j = 0; j < 4; ++j) {
            const int col = cb + (j << 4) + l;
            v2f br, bu;
            br.x = Yr[ka * CC + col];
            br.y = Yr[(ka + 1) * CC + col];
            bu.x = Yi[ka * CC + col];
            bu.y = Yi[(ka + 1) * CC + col];
            t1[j] = wmma4(fc, br, t1[j]);
            t2[j] = wmma4(fs, bu, t2[j]);
        }
    }
#pragma unroll
    for (int v = 0; v < 8; ++v) {
        const int m1 = row0 + v + (hi << 3);
        const size_t orow = ((size_t)b * TT + (size_t)(RR * m1 + m2)) * CC;
#pragma unroll
        for (int j = 0; j < 4; ++j) {
            const int cch = cb + (j << 4) + l;
            const float xv = (t1[j][v] - t2[j][v]) * invT;
            xvar[orow + cch] = xv;
            xinv[orow + cch] = x[orow + cch] - xv;
        }
    }
}

// ===========================================================================
extern "C" void kernel_launch(void* const* d_in, const int* in_sizes, int n_in,
                              void* d_out, int out_size, void* d_ws, size_t ws_size,
                              hipStream_t stream) {
    (void)in_sizes; (void)n_in; (void)out_size; (void)ws_size;

    const float* x = (const float*)d_in[0];
    float* out = (float*)d_out;
    float* xvar = out;              // [B,T,C]
    float* xinv = out + BTC;        // [B,T,C]

    // Workspace: Y re/im (128 MB) + mask (33.6 MB). The spectrum lives in
    // d_out (xvar/xinv) until inv_stage1 consumes it; inv_stage2 then
    // overwrites d_out with the final outputs.
    float* yre = (float*)d_ws;
    float* yim = yre + BTC;
    float* maskb = yim + BTC;       // [B, 2049, C]
    float* xre = xvar;              // spectrum scratch aliases d_out
    float* xim = xinv;

    const dim3 blk(256);
    fwd_stage1<<<dim3(BB * RR), blk, 0, stream>>>(x, yre, yim);
    fwd_stage2<<<dim3(BB * RR), blk, 0, stream>>>(yre, yim, xre, xim);
    mask_kernel<<<dim3(BB * CC), blk, 0, stream>>>(xre, xim, maskb);
    inv_stage1<<<dim3(BB * RR), blk, 0, stream>>>(xre, xim, maskb, yre, yim);
    inv_stage2<<<dim3(BB * RR), blk, 0, stream>>>(yre, yim, x, xvar, xinv);
}